// ProposalTargetAssigner_3925600109281
// MI455X (gfx1250) — compile-verified
//
#include <hip/hip_runtime.h>
#include <math.h>

// ---------------------------------------------------------------------------
// ProposalTargetAssigner for MI455X (gfx1250, wave32).
// 3 passes: zero gmax -> per-box global-max IoU -> assign/encode.
// Box table is staged into LDS with gfx1250 async global->LDS loads
// (ASYNCcnt / s_wait_asynccnt). Heavy rotated-IoU path is guarded by an
// exact circumradius cull and compiled ONCE (noinline) so pass2/pass3
// produce bit-identical IoU values for the force-match equality test.
// ---------------------------------------------------------------------------

#define NUM_CLASSES 3
#define NY 128
#define NX 128
#define NROT 2
#define BOX_DOF 7
#define N_BOXES 64
#define ANCH_PER_CLASS (NY * NX * NROT) /* 32768 */
#define THREADS 256
#define N_CLS_ELEMS (NUM_CLASSES * ANCH_PER_CLASS) /* 98304 */
#define BOX_DWORDS (N_BOXES * BOX_DOF) /* 448 */

struct BevBox {
  float x, y, w, l;     // center + extents (w = col3, l = col4)
  float c, s;           // cos/sin yaw
  float cx[4], cy[4];   // corners
  float area;           // w * l
  float r;              // circumradius + margin (cull only)
};

// ---- gfx1250 async global->LDS copy helpers (ASYNCcnt tracked) ------------
// LDS destination address = low 32 bits of the generic (flat) address: the
// LDS aperture keeps the workgroup-relative offset in addr[31:0].
__device__ __forceinline__ void async_ld_dword(void* lds_dst, const void* gbase,
                                               unsigned byte_off) {
  unsigned lds = (unsigned)(unsigned long long)lds_dst;
  asm volatile("global_load_async_to_lds_b32 %0, %1, %2"
               :
               : "v"(lds), "v"(byte_off), "s"(gbase)
               : "memory");
}
__device__ __forceinline__ void wait_asynccnt0() {
  asm volatile("s_wait_asynccnt 0" ::: "memory");
}

// Build BEV box from 7-DOF [x,y,z,w,l,h,yaw]. Noinline => identical codegen
// (hence bit-identical corners) in every kernel that uses it.
__device__ __attribute__((noinline)) void build_bev(const float* b7, BevBox* o) {
  float x = b7[0], y = b7[1], w = b7[3], l = b7[4], yaw = b7[6];
  float s, c;
  sincosf(yaw, &s, &c);
  o->x = x; o->y = y; o->w = w; o->l = l; o->c = c; o->s = s;
  const float hw = 0.5f * w, hl = 0.5f * l;
  const float dx[4] = { hw, -hw, -hw, hw };
  const float dy[4] = { hl,  hl, -hl, -hl };
#pragma unroll
  for (int i = 0; i < 4; ++i) {
    o->cx[i] = x + dx[i] * c - dy[i] * s;
    o->cy[i] = y + dx[i] * s + dy[i] * c;
  }
  o->area = w * l;
  o->r = 0.5f * sqrtf(w * w + l * l) + 0.01f;
}

__device__ __forceinline__ bool pt_in(const BevBox& b, float px, float py) {
  float rx = px - b.x, ry = py - b.y;
  float lx = rx * b.c + ry * b.s;
  float ly = -rx * b.s + ry * b.c;
  return (fabsf(lx) <= b.w * 0.5f + 1e-4f) && (fabsf(ly) <= b.l * 0.5f + 1e-4f);
}

// Rotated IoU exactly mirroring the reference: 24 candidates (4 corners of
// each box + 16 edge-pair intersections), mask, centroid, stable angular
// sort, shoelace. Noinline => one shared copy => bit-exact across passes.
__device__ __attribute__((noinline)) float rotated_iou(const BevBox& b1, const BevBox& b2) {
  float candx[24], candy[24];
  bool  msk[24];
#pragma unroll
  for (int i = 0; i < 4; ++i) {
    candx[i] = b1.cx[i]; candy[i] = b1.cy[i];
    msk[i]   = pt_in(b2, b1.cx[i], b1.cy[i]);
  }
#pragma unroll
  for (int j = 0; j < 4; ++j) {
    candx[4 + j] = b2.cx[j]; candy[4 + j] = b2.cy[j];
    msk[4 + j]   = pt_in(b1, b2.cx[j], b2.cy[j]);
  }
  int k = 8;
  for (int i = 0; i < 4; ++i) {
    float px = b1.cx[i], py = b1.cy[i];
    float d1x = b1.cx[(i + 1) & 3] - px, d1y = b1.cy[(i + 1) & 3] - py;
    for (int j = 0; j < 4; ++j) {
      float qx = b2.cx[j], qy = b2.cy[j];
      float d2x = b2.cx[(j + 1) & 3] - qx, d2y = b2.cy[(j + 1) & 3] - qy;
      float rqx = qx - px, rqy = qy - py;
      float den = d1x * d2y - d1y * d2x;
      float den_s = (fabsf(den) < 1e-8f) ? 1e-8f : den;
      float t = (rqx * d2y - rqy * d2x) / den_s;
      float u = (rqx * d1y - rqy * d1x) / den_s;
      bool hit = (fabsf(den) > 1e-8f) &&
                 (t >= -1e-6f) && (t <= 1.0f + 1e-6f) &&
                 (u >= -1e-6f) && (u <= 1.0f + 1e-6f);
      candx[k] = px + t * d1x;
      candy[k] = py + t * d1y;
      msk[k] = hit;
      ++k;
    }
  }
  // Compact valid candidates (order preserved => stable-sort equivalence).
  float ptsx[24], ptsy[24];
  int cnt = 0;
  float sx = 0.0f, sy = 0.0f;
  for (int i = 0; i < 24; ++i) {
    if (msk[i]) {
      ptsx[cnt] = candx[i]; ptsy[cnt] = candy[i];
      sx += candx[i]; sy += candy[i];
      ++cnt;
    }
  }
  float inter = 0.0f;
  if (cnt > 0) {
    float inv = 1.0f / (float)cnt;
    float ctrx = sx * inv, ctry = sy * inv;
    float ang[24];
    for (int i = 0; i < cnt; ++i)
      ang[i] = atan2f(ptsy[i] - ctry, ptsx[i] - ctrx);
    // Stable insertion sort by angle (invalid pts were already dropped; in
    // the reference they sort last with ang=1e9 and collapse onto the last
    // valid point, contributing 0 to the shoelace sum).
    for (int i = 1; i < cnt; ++i) {
      float a = ang[i], xx = ptsx[i], yy = ptsy[i];
      int j = i - 1;
      while (j >= 0 && ang[j] > a) {
        ang[j + 1] = ang[j]; ptsx[j + 1] = ptsx[j]; ptsy[j + 1] = ptsy[j];
        --j;
      }
      ang[j + 1] = a; ptsx[j + 1] = xx; ptsy[j + 1] = yy;
    }
    float acc = 0.0f;
    for (int i = 0; i < cnt; ++i) {
      int n = (i + 1 == cnt) ? 0 : i + 1;
      acc += ptsx[i] * ptsy[n] - ptsy[i] * ptsx[n];
    }
    inter = 0.5f * fabsf(acc);
  }
  float uni = b1.area + b2.area - inter;
  return inter / fmaxf(uni, 1e-7f);
}

// ---------------------------------------------------------------------------
__global__ void zero_gmax_kernel(unsigned int* g) {
  int t = blockIdx.x * blockDim.x + threadIdx.x;
  if (t < NUM_CLASSES * N_BOXES) g[t] = 0u;
}

// Pass 2: gmax[c][b] = max over anchors of class-masked IoU (bit-pattern max;
// IoU >= 0 so uint ordering == float ordering).
__global__ void gmax_kernel(const float* __restrict__ boxes,
                            const int* __restrict__ class_idx,
                            const float* __restrict__ anchors,
                            unsigned int* __restrict__ gmaxG) {
  __shared__ float  sRaw[BOX_DWORDS];      // raw 64x7 box table
  __shared__ int    sCls[N_BOXES];
  __shared__ BevBox sBox[N_BOXES];
  __shared__ unsigned int sG[N_BOXES];

  const unsigned t = threadIdx.x;

  // Async-stage the shared tables: ASYNCcnt-tracked global->LDS copies
  // are drained before the barrier; anchor load/build starts after it.
  if (t < BOX_DWORDS)          async_ld_dword(&sRaw[t],       boxes,     t * 4u);
  if (t + 256u < BOX_DWORDS)   async_ld_dword(&sRaw[t + 256], boxes,     (t + 256u) * 4u);
  if (t < N_BOXES)             async_ld_dword(&sCls[t],       class_idx, t * 4u);
  if (t < N_BOXES)             sG[t] = 0u;
  wait_asynccnt0();            // each wave drains its own async loads
  __syncthreads();             // ... then the barrier publishes all LDS writes

  if (t < N_BOXES) build_bev(&sRaw[t * BOX_DOF], &sBox[t]);
  __syncthreads();

  const int c = blockIdx.y;
  const int aidx = blockIdx.x * blockDim.x + t;
  const size_t idx = (size_t)c * ANCH_PER_CLASS + aidx;
  const float* A = anchors + idx * BOX_DOF;
  __builtin_prefetch(A, 0, 1);             // gfx1250 global_prefetch_b8

  BevBox anc;
  build_bev(A, &anc);

  for (int b = 0; b < N_BOXES; ++b) {
    if (sCls[b] != c) continue;
    float dx = sBox[b].x - anc.x, dy = sBox[b].y - anc.y;
    float rr = sBox[b].r + anc.r;
    if (dx * dx + dy * dy > rr * rr) continue;   // provably disjoint -> IoU==0
    float v = rotated_iou(sBox[b], anc);
    if (v > 0.0f) atomicMax(&sG[b], __float_as_uint(v));
  }
  __syncthreads();
  if (t < N_BOXES) {
    unsigned int v = sG[t];
    if (v) atomicMax(&gmaxG[c * N_BOXES + t], v);
  }
}

// Pass 3: per-anchor amax/argmax/force, labels, and regression encoding.
__global__ void assign_kernel(const float* __restrict__ boxes,
                              const int* __restrict__ class_idx,
                              const float* __restrict__ anchors,
                              const unsigned int* __restrict__ gmaxG,
                              float* __restrict__ out) {
  __shared__ float  sRaw[BOX_DWORDS];      // raw 64x7 box table (also encode src)
  __shared__ int    sCls[N_BOXES];
  __shared__ BevBox sBox[N_BOXES];
  __shared__ unsigned int sGm[N_BOXES];    // per-box global max for this class

  const unsigned t = threadIdx.x;
  const int c = blockIdx.y;

  if (t < BOX_DWORDS)          async_ld_dword(&sRaw[t],       boxes,     t * 4u);
  if (t + 256u < BOX_DWORDS)   async_ld_dword(&sRaw[t + 256], boxes,     (t + 256u) * 4u);
  if (t < N_BOXES)             async_ld_dword(&sCls[t],       class_idx, t * 4u);
  if (t < N_BOXES)             async_ld_dword(&sGm[t],        gmaxG,
                                              (unsigned)(c * N_BOXES + t) * 4u);
  wait_asynccnt0();
  __syncthreads();

  if (t < N_BOXES) build_bev(&sRaw[t * BOX_DOF], &sBox[t]);
  __syncthreads();

  const int aidx = blockIdx.x * blockDim.x + t;
  const size_t idx = (size_t)c * ANCH_PER_CLASS + aidx;
  const float* A = anchors + idx * BOX_DOF;
  __builtin_prefetch(A, 0, 1);             // gfx1250 global_prefetch_b8

  BevBox anc;
  build_bev(A, &anc);

  float best = -1.0f;
  int bestb = 0;
  bool force = false;
  for (int b = 0; b < N_BOXES; ++b) {
    float v = 0.0f;                        // class-mismatch / disjoint -> 0
    if (sCls[b] == c) {
      float dx = sBox[b].x - anc.x, dy = sBox[b].y - anc.y;
      float rr = sBox[b].r + anc.r;
      if (dx * dx + dy * dy <= rr * rr) v = rotated_iou(sBox[b], anc);
    }
    if (v > best) { best = v; bestb = b; } // strict > == first-occurrence argmax
    float g = __uint_as_float(sGm[b]);
    force = force || ((g > 0.0f) && (v == g));
  }

  const float hi = (c == 0) ? 0.6f  : 0.5f;
  const float lo = (c == 0) ? 0.45f : 0.35f;
  int lab = force ? 1 : ((best >= hi) ? 1 : ((best < lo) ? 0 : -1));

  const float gcls = (lab == 1) ? 1.0f : 0.0f;   // max(lab, 0)
  const float mcls = (lab != -1) ? 1.0f : 0.0f;  // lab != -1
  const bool  mreg = (lab == 1);

  // Output layout: [G_cls | M_cls | G_reg | M_reg], all flattened row-major.
  out[idx] = gcls;
  out[(size_t)N_CLS_ELEMS + idx] = mcls;

  const size_t regBase = (size_t)2 * N_CLS_ELEMS + idx * BOX_DOF;
  if (mreg) {
    const float* g = &sRaw[bestb * BOX_DOF];
    float d = sqrtf(A[3] * A[3] + A[4] * A[4]);
    out[regBase + 0] = (g[0] - A[0]) / d;
    out[regBase + 1] = (g[1] - A[1]) / d;
    out[regBase + 2] = (g[2] - A[2]) / A[5];
    out[regBase + 3] = logf(g[3] / A[3]);
    out[regBase + 4] = logf(g[4] / A[4]);
    out[regBase + 5] = logf(g[5] / A[5]);
    out[regBase + 6] = g[6] - A[6];
  } else {
#pragma unroll
    for (int i = 0; i < BOX_DOF; ++i) out[regBase + i] = 0.0f;
  }

  out[(size_t)2 * N_CLS_ELEMS + (size_t)N_CLS_ELEMS * BOX_DOF + idx] = mreg ? 1.0f : 0.0f;
}

// ---------------------------------------------------------------------------
extern "C" void kernel_launch(void* const* d_in, const int* in_sizes, int n_in,
                              void* d_out, int out_size, void* d_ws, size_t ws_size,
                              hipStream_t stream) {
  (void)in_sizes; (void)n_in; (void)out_size; (void)ws_size;
  const float* boxes     = (const float*)d_in[0];
  const int*   class_idx = (const int*)d_in[1];
  /* d_in[2] = box_ignore: unused by the reference computation */
  const float* anchors   = (const float*)d_in[3];
  unsigned int* gmaxG    = (unsigned int*)d_ws;   // 192 uints
  float* out             = (float*)d_out;

  zero_gmax_kernel<<<1, 256, 0, stream>>>(gmaxG);

  dim3 grid(ANCH_PER_CLASS / THREADS, NUM_CLASSES, 1);
  gmax_kernel<<<grid, THREADS, 0, stream>>>(boxes, class_idx, anchors, gmaxG);
  assign_kernel<<<grid, THREADS, 0, stream>>>(boxes, class_idx, anchors, gmaxG, out);
}